// CRF_Loss_13305808683352
// MI455X (gfx1250) — compile-verified
//
#include <hip/hip_runtime.h>
#include <math.h>

// ---------------- problem constants (from reference) ----------------
#define NB 512      // batch
#define NS 1024     // sequence length
#define NT 50       // tags
#define MB 16       // batches per forward block
#define AST 67      // LDS row stride for aExp (67 mod 64 = 3 -> conflict-free A reads)
#define NK 13       // K-chunks of 4 actually needed: 13*4 = 52 >= NT (rest provably zero)

typedef __attribute__((ext_vector_type(2))) float v2f;
typedef __attribute__((ext_vector_type(8))) float v8f;

// =====================================================================
// Forward algorithm in exp-space via V_WMMA_F32_16X16X4_F32.
// Block = 16 batches, 4 waves; wave w owns output columns [16w, 16w+16).
// Per step: D(16x16) = A(16x52 aExp) @ B(52x16 expTrans) per wave,
// then log/renormalize/exp back into LDS for the next step.
// =====================================================================
__global__ __launch_bounds__(128) void crf_forward_kernel(
    const float* __restrict__ tok,     // [NB, NS, NT]
    const int*   __restrict__ mask,    // [NB, NS]
    const float* __restrict__ trans,   // [NT, NT]
    const float* __restrict__ startT,  // [NT]
    const float* __restrict__ endT,    // [NT]
    float* __restrict__ den)           // [NB] log-partition per batch
{
  __shared__ float aExp[MB][AST];   // exp(alpha - rowOff), padded cols = 0
  __shared__ float part[4][MB];     // per-wave row statistics
  __shared__ float rowOff[MB];      // running log normalizer
  __shared__ int   lenSh[MB];       // sequence lengths

  const int tid   = threadIdx.x;
  const int w     = tid >> 5;       // wave id 0..3
  const int lane  = tid & 31;
  const int hl    = lane & 15;      // lane within half-wave
  const int hi    = lane >> 4;      // half select
  const int col   = w * 16 + hl;    // output column (N) this lane owns
  const int colC  = (col < NT) ? col : 0;   // clamped column for branch-free loads
  const bool colOK = (col < NT);
  const int rBase = hi * 8;         // D-matrix row base for this half
  const int aOff  = hi * 2;         // A/B K offset for this half
  const int bBase = blockIdx.x * MB;

  // ---- sequence lengths: len[b] = sum_t mask[b][t] ----
  if (tid < MB) lenSh[tid] = 0;
  __syncthreads();
  {
    int r = tid >> 3;               // 8 threads per batch row
    int sub = tid & 7;
    int acc = 0;
    for (int t = sub; t < NS; t += 8) acc += mask[(bBase + r) * NS + t];
    atomicAdd(&lenSh[r], acc);
  }

  // ---- B-matrix registers: exp(transitions), zero-padded, loaded once ----
  // Layout mirrors A: vgpr .x holds K=4k+aOff, .y holds K=4k+aOff+1.
  v2f Breg[NK];
#pragma unroll
  for (int k = 0; k < NK; ++k) {
    int i0 = 4 * k + aOff;
    int i1 = i0 + 1;
    float b0 = (i0 < NT && colOK) ? expf(trans[i0 * NT + col]) : 0.0f;
    float b1 = (i1 < NT && colOK) ? expf(trans[i1 * NT + col]) : 0.0f;
    Breg[k].x = b0;
    Breg[k].y = b1;
  }
  const float eEnd = colOK ? expf(endT[col]) : 0.0f;

  __syncthreads();

  float rel[8];

  // ---- t = 0: alpha0 = start + token_scores[:,0] ----
#pragma unroll
  for (int v = 0; v < 8; ++v) {
    int r = rBase + v;
    float tv = startT[colC] + tok[(bBase + r) * NS * NT + colC];
    float a0 = colOK ? tv : -INFINITY;
    rel[v] = a0;
    float m = a0;
#pragma unroll
    for (int d = 1; d < 16; d <<= 1) m = fmaxf(m, __shfl_xor(m, d, 16));
    if (hl == v) part[w][r] = m;
  }
  __syncthreads();
#pragma unroll
  for (int v = 0; v < 8; ++v) {
    int r = rBase + v;
    float nm = fmaxf(fmaxf(part[0][r], part[1][r]),
                     fmaxf(part[2][r], part[3][r]));
    aExp[r][col] = expf(rel[v] - nm);      // exp(-inf - nm) = 0 for padded cols
    if (w == 0 && hl == v) rowOff[r] = nm;
  }
  __syncthreads();

  // ---- sequential scan over steps 1..NS-1 ----
  for (int t = 1; t < NS; ++t) {
    // Emission loads issued first (branch-free, clamped) so their HBM
    // latency hides under the WMMA chain below.
    float tv[8];
#pragma unroll
    for (int v = 0; v < 8; ++v) {
      int r = rBase + v;
      tv[v] = tok[(bBase + r) * NS * NT + t * NT + colC];
    }
    {
      int tn = (t + 1 < NS) ? (t + 1) : t;
      __builtin_prefetch(&tok[(bBase + rBase) * NS * NT + tn * NT + colC], 0, 1);
    }

    // A-phase: two independent WMMA accumulation chains over K=0..51
    v8f acc0 = {};
    v8f acc1 = {};
#pragma unroll
    for (int k = 0; k < NK - 1; k += 2) {
      v2f A0, A1;
      A0.x = aExp[hl][4 * k + aOff];
      A0.y = aExp[hl][4 * k + aOff + 1];
      A1.x = aExp[hl][4 * (k + 1) + aOff];
      A1.y = aExp[hl][4 * (k + 1) + aOff + 1];
      acc0 = __builtin_amdgcn_wmma_f32_16x16x4_f32(
          false, A0, false, Breg[k], (short)0, acc0, false, false);
      acc1 = __builtin_amdgcn_wmma_f32_16x16x4_f32(
          false, A1, false, Breg[k + 1], (short)0, acc1, false, false);
    }
    {
      v2f A12;
      A12.x = aExp[hl][4 * (NK - 1) + aOff];
      A12.y = aExp[hl][4 * (NK - 1) + aOff + 1];
      acc0 = __builtin_amdgcn_wmma_f32_16x16x4_f32(
          false, A12, false, Breg[NK - 1], (short)0, acc0, false, false);
    }

    // B-phase: back to log space, add emissions, per-row max over this wave
#pragma unroll
    for (int v = 0; v < 8; ++v) {
      int r = rBase + v;
      float s  = acc0[v] + acc1[v];
      float rl = logf(s) + (colOK ? tv[v] : -INFINITY);  // log(0) = -inf pads
      rel[v] = rl;
      float m = rl;
#pragma unroll
      for (int d = 1; d < 16; d <<= 1) m = fmaxf(m, __shfl_xor(m, d, 16));
      if (hl == v) part[w][r] = m;
    }
    __syncthreads();

    // C-phase: combine row max across waves, renormalize, masked write-back
#pragma unroll
    for (int v = 0; v < 8; ++v) {
      int r = rBase + v;
      float nm = fmaxf(fmaxf(part[0][r], part[1][r]),
                       fmaxf(part[2][r], part[3][r]));
      bool active = (t < lenSh[r]);          // mask[b][t] == (t < len[b])
      float ae = expf(rel[v] - nm);
      if (active) {
        aExp[r][col] = ae;
        if (w == 0 && hl == v) rowOff[r] += nm;
      }
    }
    __syncthreads();
  }

  // ---- denominator: rowOff + log(sum_j aExp * exp(end)) ----
#pragma unroll
  for (int v = 0; v < 8; ++v) {
    int r = rBase + v;
    float s = aExp[r][col] * eEnd;
#pragma unroll
    for (int d = 1; d < 16; d <<= 1) s += __shfl_xor(s, d, 16);
    if (hl == v) part[w][r] = s;
  }
  __syncthreads();
  if (w == 0 && lane < MB) {
    float tot = part[0][lane] + part[1][lane] + part[2][lane] + part[3][lane];
    den[bBase + lane] = rowOff[lane] + logf(tot);
  }
}

// =====================================================================
// Numerator: start + sum(emit*mask) + sum(trans[tag_{t-1},tag_t]*mask[1:])
//            + end[last_tag].  One block per batch, fixed-order reduction.
// =====================================================================
__global__ __launch_bounds__(256) void crf_numer_kernel(
    const float*     __restrict__ tok,    // [NB, NS, NT]
    const long long* __restrict__ tags,   // [NB, NS] int64
    const int*       __restrict__ mask,   // [NB, NS]
    const float*     __restrict__ trans,  // [NT, NT]
    const float*     __restrict__ startT, // [NT]
    const float*     __restrict__ endT,   // [NT]
    float* __restrict__ num)              // [NB]
{
  __shared__ float red[256];
  const int b = blockIdx.x;
  const int tid = threadIdx.x;

  float p = 0.0f;
  for (int t = tid; t < NS; t += 256) {
    int m = mask[b * NS + t];
    if (m) {
      int tg = (int)tags[b * NS + t];
      p += tok[(b * NS + t) * NT + tg];                       // emission
      if (t >= 1) {
        int pg = (int)tags[b * NS + t - 1];
        p += trans[pg * NT + tg];                             // transition
      }
      if (t == NS - 1 || mask[b * NS + t + 1] == 0)
        p += endT[tg];                                        // end (last valid)
    }
    if (t == 0) p += startT[(int)tags[b * NS]];               // start
  }
  red[tid] = p;
  __syncthreads();
  for (int s = 128; s > 0; s >>= 1) {
    if (tid < s) red[tid] += red[tid + s];
    __syncthreads();
  }
  if (tid == 0) num[b] = red[0];
}

// =====================================================================
// Final: out = sum_b(den - num) / NB   (fixed-order -> deterministic)
// =====================================================================
__global__ __launch_bounds__(512) void crf_reduce_kernel(
    const float* __restrict__ den, const float* __restrict__ num,
    float* __restrict__ out)
{
  __shared__ float red[512];
  const int tid = threadIdx.x;
  red[tid] = den[tid] - num[tid];
  __syncthreads();
  for (int s = 256; s > 0; s >>= 1) {
    if (tid < s) red[tid] += red[tid + s];
    __syncthreads();
  }
  if (tid == 0) out[0] = red[0] / (float)NB;
}

extern "C" void kernel_launch(void* const* d_in, const int* in_sizes, int n_in,
                              void* d_out, int out_size, void* d_ws, size_t ws_size,
                              hipStream_t stream) {
  (void)in_sizes; (void)n_in; (void)out_size; (void)ws_size;
  const float*     tok    = (const float*)d_in[0];      // f32 [B,S,T]
  const long long* tags   = (const long long*)d_in[1];  // int64 [B,S]
  const int*       mask   = (const int*)d_in[2];        // int32 [B,S]
  const float*     trans  = (const float*)d_in[3];      // f32 [T,T]
  const float*     startT = (const float*)d_in[4];      // f32 [T]
  const float*     endT   = (const float*)d_in[5];      // f32 [T]
  float* out = (float*)d_out;
  float* den = (float*)d_ws;        // [NB]
  float* num = den + NB;            // [NB]

  crf_forward_kernel<<<NB / MB, 128, 0, stream>>>(tok, mask, trans, startT, endT, den);
  crf_numer_kernel<<<NB, 256, 0, stream>>>(tok, tags, mask, trans, startT, endT, num);
  crf_reduce_kernel<<<1, 512, 0, stream>>>(den, num, out);
}